// GNNModule_33054068310342
// MI455X (gfx1250) — compile-verified
//
#include <hip/hip_runtime.h>

typedef __attribute__((ext_vector_type(2))) float v2f;
typedef __attribute__((ext_vector_type(8))) float v8f;

#define N_NODES  100000
#define N_EDGES  800000
#define N_GRAPHS 128
#define D_IN     84
#define D_HID    840
#define D_FC1    1024
#define D_OUT    384

// ---------------------------------------------------------------------------
// Scatter-add over edges: agg[dst[e]][f] += h[src[e]][f]
// ---------------------------------------------------------------------------
__global__ void __launch_bounds__(256) scatter_add_kernel(
    const float* __restrict__ h, const int* __restrict__ src,
    const int* __restrict__ dst, float* __restrict__ agg,
    int nEdges, int D)
{
    long long idx = (long long)blockIdx.x * blockDim.x + threadIdx.x;
    long long total = (long long)nEdges * D;
    if (idx >= total) return;
    int e = (int)(idx / D);
    int f = (int)(idx % D);
    int s = src[e];
    int d = dst[e];
    atomicAdd(&agg[(size_t)d * D + f], h[(size_t)s * D + f]);
}

// ---------------------------------------------------------------------------
// Fused GEMM: out = act((A0 [+ A1]) @ W + bias), via V_WMMA_F32_16X16X4_F32.
// N, K are compile-time so W loads use immediate byte offsets (k0*N*4 fits the
// 24-bit IOFFSET at every call site) and there is no 64-bit pointer math in
// the K loop. One wave32 computes TWO vertically-adjacent 16x16 tiles,
// sharing each B fragment across two WMMAs. The loop is software-pipelined:
// fragments for step k+1 are loaded while the WMMAs of step k execute.
// Requirements: M % 32 == 0, K % 4 == 0 (true everywhere). Ragged N handled
// by clamping B/bias column addresses: out-of-range lanes only feed output
// columns that are never stored.
//
// f32 16x4 A fragment (ISA 7.12.2): lanes 0-15 -> M=lane, v0:K=k0, v1:K=k0+1
//                                   lanes16-31 -> M=lane-16, K=k0+2 / k0+3
// f32 4x16 B fragment: v0: lanes 0-15 K=k0, lanes16-31 K=k0+2; v1: +1 ; N=lane&15
// f32 16x16 C/D: VGPR i, lanes 0-15 -> M=i, lanes16-31 -> M=i+8 ; N=lane&15
// ---------------------------------------------------------------------------
template <bool HAS_A1, int N, int K>
__global__ void __launch_bounds__(256) gemm_wmma_bias_act(
    const float* __restrict__ A0, const float* __restrict__ A1,
    const float* __restrict__ W,  const float* __restrict__ bias,
    float* __restrict__ out, int totalTiles, int doRelu)
{
    constexpr int nTiles = (N + 15) / 16;

    int wave = (int)(blockIdx.x * (blockDim.x >> 5) + (threadIdx.x >> 5));
    if (wave >= totalTiles) return;           // wave-uniform exit
    int mt2 = wave / nTiles;                  // m-tile PAIR index (32 rows)
    int nt  = wave % nTiles;

    int lane = threadIdx.x & 31;
    int half = lane >> 4;                     // K pair {0,1} vs {2,3}
    int l    = lane & 15;

    int row0 = mt2 * 32 + l;
    int row1 = row0 + 16;
    int col  = nt * 16 + l;
    int colC = (col < N) ? col : (N - 1);     // clamped address, never stored

    const float* a0p = A0 + (size_t)row0 * K + 2 * half;
    const float* a1p = A0 + (size_t)row1 * K + 2 * half;
    const float* g0p = HAS_A1 ? (A1 + (size_t)row0 * K + 2 * half) : nullptr;
    const float* g1p = HAS_A1 ? (A1 + (size_t)row1 * K + 2 * half) : nullptr;
    const float* wp  = W + (size_t)(2 * half) * N + colC;

    auto loadFrag = [&](int k0, v2f& a0, v2f& a1, v2f& b) {
        a0.x = a0p[k0];     a0.y = a0p[k0 + 1];
        a1.x = a1p[k0];     a1.y = a1p[k0 + 1];
        if constexpr (HAS_A1) {
            a0.x += g0p[k0]; a0.y += g0p[k0 + 1];
            a1.x += g1p[k0]; a1.y += g1p[k0 + 1];
        }
        b.x = wp[(size_t)k0 * N];
        b.y = wp[(size_t)k0 * N + N];
    };

    v8f acc0 = {};
    v8f acc1 = {};

    v2f a0c, a1c, bc;
    loadFrag(0, a0c, a1c, bc);
#pragma unroll 4
    for (int k0 = 4; k0 < K; k0 += 4) {
        v2f a0n, a1n, bn;
        loadFrag(k0, a0n, a1n, bn);           // next step's loads in flight...
        acc0 = __builtin_amdgcn_wmma_f32_16x16x4_f32(
            false, a0c, false, bc, (short)0, acc0, false, false);
        acc1 = __builtin_amdgcn_wmma_f32_16x16x4_f32(
            false, a1c, false, bc, (short)0, acc1, false, false);
        a0c = a0n; a1c = a1n; bc = bn;
    }
    acc0 = __builtin_amdgcn_wmma_f32_16x16x4_f32(
        false, a0c, false, bc, (short)0, acc0, false, false);
    acc1 = __builtin_amdgcn_wmma_f32_16x16x4_f32(
        false, a1c, false, bc, (short)0, acc1, false, false);

    if (col < N) {
        float bv = bias[col];
#pragma unroll
        for (int i = 0; i < 8; ++i) {
            int r0 = mt2 * 32 + i + half * 8;
            float v0 = acc0[i] + bv;
            float v1 = acc1[i] + bv;
            if (doRelu) { v0 = fmaxf(v0, 0.0f); v1 = fmaxf(v1, 0.0f); }
            out[(size_t)r0 * N + col]        = v0;
            out[(size_t)(r0 + 16) * N + col] = v1;
        }
    }
}

// ---------------------------------------------------------------------------
// Per-graph node counts
// ---------------------------------------------------------------------------
__global__ void __launch_bounds__(256) count_kernel(
    const int* __restrict__ batch, int* __restrict__ counts, int nNodes)
{
    int i = blockIdx.x * blockDim.x + threadIdx.x;
    if (i < nNodes) atomicAdd(&counts[batch[i]], 1);
}

// ---------------------------------------------------------------------------
// Segment sum + segment max (h >= 0 post-ReLU, so unsigned-bit atomicMax with
// zero init is exact and matches where(count>0, segmax, 0)).
// ---------------------------------------------------------------------------
__global__ void __launch_bounds__(256) pool_kernel(
    const float* __restrict__ h, const int* __restrict__ batch,
    float* __restrict__ sums, unsigned* __restrict__ maxbits,
    int nNodes, int D)
{
    long long idx = (long long)blockIdx.x * blockDim.x + threadIdx.x;
    long long total = (long long)nNodes * D;
    if (idx >= total) return;
    int node = (int)(idx / D);
    int f    = (int)(idx % D);
    int g = batch[node];
    float v = h[(size_t)node * D + f];
    atomicAdd(&sums[(size_t)g * D + f], v);
    atomicMax(&maxbits[(size_t)g * D + f], __float_as_uint(v));
}

// ---------------------------------------------------------------------------
// z[g] = concat(maxp[g], sums[g]/max(count,1))   -> [G, 2*D]
// ---------------------------------------------------------------------------
__global__ void __launch_bounds__(256) build_z_kernel(
    const unsigned* __restrict__ maxbits, const float* __restrict__ sums,
    const int* __restrict__ counts, float* __restrict__ z, int G, int D)
{
    int idx = blockIdx.x * blockDim.x + threadIdx.x;
    if (idx >= G * D) return;
    int g = idx / D;
    int f = idx % D;
    float cnt = (float)counts[g];
    z[(size_t)g * (2 * D) + f]     = __uint_as_float(maxbits[(size_t)g * D + f]);
    z[(size_t)g * (2 * D) + D + f] = sums[(size_t)g * D + f] / fmaxf(cnt, 1.0f);
}

// ---------------------------------------------------------------------------
// Launch
// ---------------------------------------------------------------------------
extern "C" void kernel_launch(void* const* d_in, const int* in_sizes, int n_in,
                              void* d_out, int out_size, void* d_ws, size_t ws_size,
                              hipStream_t stream)
{
    (void)in_sizes; (void)n_in; (void)out_size; (void)ws_size;

    const float* x    = (const float*)d_in[0];
    const int*   ei   = (const int*)d_in[1];        // [2, N_EDGES] flat
    const int*   batch= (const int*)d_in[2];
    const float* W1   = (const float*)d_in[3];
    const float* b1   = (const float*)d_in[4];
    const float* W2   = (const float*)d_in[5];
    const float* b2   = (const float*)d_in[6];
    const float* Wg1  = (const float*)d_in[7];
    const float* bg1  = (const float*)d_in[8];
    const float* Wg2  = (const float*)d_in[9];
    const float* bg2  = (const float*)d_in[10];
    const int* src = ei;
    const int* dst = ei + N_EDGES;

    // Workspace layout (bytes)
    char* ws = (char*)d_ws;
    size_t off = 0;
    float*    agg     = (float*)(ws + off); off += (size_t)N_NODES * D_IN  * 4;  // 33.6 MB
    float*    h1      = (float*)(ws + off); off += (size_t)N_NODES * D_IN  * 4;  // 33.6 MB
    float*    h2      = (float*)(ws + off); off += (size_t)N_NODES * D_HID * 4;  // 336 MB
    float*    sums    = (float*)(ws + off); off += (size_t)N_GRAPHS * D_HID * 4;
    unsigned* maxbits = (unsigned*)(ws + off); off += (size_t)N_GRAPHS * D_HID * 4;
    int*      counts  = (int*)(ws + off); off += (size_t)N_GRAPHS * 4;
    float*    z       = (float*)(ws + off); off += (size_t)N_GRAPHS * 2 * D_HID * 4;
    float*    fc1     = (float*)(ws + off); off += (size_t)N_GRAPHS * D_FC1 * 4;

    float* out = (float*)d_out;

    const int TPB = 256;
    const int WPB = TPB / 32;   // waves per block

    // ---- conv1: agg = scatter(x); h1 = relu((x+agg) @ W1 + b1) -------------
    hipMemsetAsync(agg, 0, (size_t)N_NODES * D_IN * 4, stream);
    {
        long long tot = (long long)N_EDGES * D_IN;
        int blocks = (int)((tot + TPB - 1) / TPB);
        scatter_add_kernel<<<blocks, TPB, 0, stream>>>(x, src, dst, agg, N_EDGES, D_IN);
    }
    {
        int mPairs = N_NODES / 32;                 // 3125 (exact)
        int nTiles = (D_IN + 15) / 16;             // 6
        int total  = mPairs * nTiles;
        int blocks = (total + WPB - 1) / WPB;
        gemm_wmma_bias_act<true, D_IN, D_IN><<<blocks, TPB, 0, stream>>>(
            x, agg, W1, b1, h1, total, 1);
    }

    // ---- conv2: agg = scatter(h1); h2 = relu((h1+agg) @ W2 + b2) -----------
    hipMemsetAsync(agg, 0, (size_t)N_NODES * D_IN * 4, stream);
    {
        long long tot = (long long)N_EDGES * D_IN;
        int blocks = (int)((tot + TPB - 1) / TPB);
        scatter_add_kernel<<<blocks, TPB, 0, stream>>>(h1, src, dst, agg, N_EDGES, D_IN);
    }
    {
        int mPairs = N_NODES / 32;                 // 3125
        int nTiles = (D_HID + 15) / 16;            // 53 (840 = 52*16+8, ragged)
        int total  = mPairs * nTiles;
        int blocks = (total + WPB - 1) / WPB;
        gemm_wmma_bias_act<true, D_HID, D_IN><<<blocks, TPB, 0, stream>>>(
            h1, agg, W2, b2, h2, total, 1);
    }

    // ---- pooling -----------------------------------------------------------
    hipMemsetAsync(sums,    0, (size_t)N_GRAPHS * D_HID * 4, stream);
    hipMemsetAsync(maxbits, 0, (size_t)N_GRAPHS * D_HID * 4, stream);
    hipMemsetAsync(counts,  0, (size_t)N_GRAPHS * 4, stream);
    {
        int blocks = (N_NODES + TPB - 1) / TPB;
        count_kernel<<<blocks, TPB, 0, stream>>>(batch, counts, N_NODES);
    }
    {
        long long tot = (long long)N_NODES * D_HID;
        int blocks = (int)((tot + TPB - 1) / TPB);
        pool_kernel<<<blocks, TPB, 0, stream>>>(h2, batch, sums, maxbits, N_NODES, D_HID);
    }
    {
        int tot = N_GRAPHS * D_HID;
        int blocks = (tot + TPB - 1) / TPB;
        build_z_kernel<<<blocks, TPB, 0, stream>>>(maxbits, sums, counts, z, N_GRAPHS, D_HID);
    }

    // ---- fc_g[0]: fc1 = relu(z @ Wg1 + bg1)  [128,1680]x[1680,1024] --------
    {
        int mPairs = N_GRAPHS / 32;                // 4
        int nTiles = D_FC1 / 16;                   // 64 (exact)
        int total  = mPairs * nTiles;
        int blocks = (total + WPB - 1) / WPB;
        gemm_wmma_bias_act<false, D_FC1, 2 * D_HID><<<blocks, TPB, 0, stream>>>(
            z, nullptr, Wg1, bg1, fc1, total, 1);
    }

    // ---- fc_g[3]: out = fc1 @ Wg2 + bg2  [128,1024]x[1024,384] -------------
    {
        int mPairs = N_GRAPHS / 32;                // 4
        int nTiles = D_OUT / 16;                   // 24 (exact)
        int total  = mPairs * nTiles;
        int blocks = (total + WPB - 1) / WPB;
        gemm_wmma_bias_act<false, D_OUT, D_FC1><<<blocks, TPB, 0, stream>>>(
            fc1, nullptr, Wg2, bg2, out, total, 0);
    }
}